// SSR_71279277244826
// MI455X (gfx1250) — compile-verified
//
#include <hip/hip_runtime.h>
#include <math.h>

// Problem constants (from the reference: images (8,3,300,64,64), fps=30)
#define B_DIM 8
#define K_DIM 300
#define N_PIX 4096          // 64*64
#define L_WIN 30            // fps
#define W_DIM (K_DIM - L_WIN)  // 270

typedef __attribute__((ext_vector_type(2))) float v2f;
typedef __attribute__((ext_vector_type(8))) float v8f;

__device__ __forceinline__ float wave_sum32(float x) {
#pragma unroll
  for (int off = 16; off > 0; off >>= 1) x += __shfl_xor(x, off, 32);
  return x;
}

// ---------------------------------------------------------------------------
// Kernel 0: zero the output (harness poisons d_out; scatter-add needs zeros)
// ---------------------------------------------------------------------------
__global__ void zero_kernel(float* __restrict__ out, int n) {
  int i = blockIdx.x * blockDim.x + threadIdx.x;
  if (i < n) out[i] = 0.f;
}

// ---------------------------------------------------------------------------
// Kernel 1: per-frame 3x3 covariance  Cm[b,k] = V^T V / N   (V = 4096 x 3)
// Memory-bound: streams 118 MB once.  One block (256 thr = 8 waves) per frame.
// images layout: (B, C=3, K, 64, 64) -> component planes are contiguous 4096f.
// ---------------------------------------------------------------------------
__global__ void cov_kernel(const float* __restrict__ img, float* __restrict__ cov) {
  const int fk = blockIdx.x;              // b*K_DIM + k
  const int b = fk / K_DIM, k = fk % K_DIM;
  const size_t plane = (size_t)K_DIM * N_PIX;
  const float4* __restrict__ px =
      (const float4*)(img + ((size_t)(b * 3 + 0)) * plane + (size_t)k * N_PIX);
  const float4* __restrict__ py =
      (const float4*)(img + ((size_t)(b * 3 + 1)) * plane + (size_t)k * N_PIX);
  const float4* __restrict__ pz =
      (const float4*)(img + ((size_t)(b * 3 + 2)) * plane + (size_t)k * N_PIX);

  float sxx = 0.f, sxy = 0.f, sxz = 0.f, syy = 0.f, syz = 0.f, szz = 0.f;
#pragma unroll
  for (int it = 0; it < 4; ++it) {
    int i = it * 256 + threadIdx.x;       // 1024 float4 per frame plane
    float4 x = px[i], y = py[i], z = pz[i];
    sxx += x.x * x.x + x.y * x.y + x.z * x.z + x.w * x.w;
    sxy += x.x * y.x + x.y * y.y + x.z * y.z + x.w * y.w;
    sxz += x.x * z.x + x.y * z.y + x.z * z.z + x.w * z.w;
    syy += y.x * y.x + y.y * y.y + y.z * y.z + y.w * y.w;
    syz += y.x * z.x + y.y * z.y + y.z * z.z + y.w * z.w;
    szz += z.x * z.x + z.y * z.y + z.z * z.z + z.w * z.w;
  }
  sxx = wave_sum32(sxx); sxy = wave_sum32(sxy); sxz = wave_sum32(sxz);
  syy = wave_sum32(syy); syz = wave_sum32(syz); szz = wave_sum32(szz);

  __shared__ float sh[8][6];
  const int wave = threadIdx.x >> 5, lane = threadIdx.x & 31;
  if (lane == 0) {
    sh[wave][0] = sxx; sh[wave][1] = sxy; sh[wave][2] = sxz;
    sh[wave][3] = syy; sh[wave][4] = syz; sh[wave][5] = szz;
  }
  __syncthreads();
  if (threadIdx.x < 6) {
    float t = 0.f;
#pragma unroll
    for (int w = 0; w < 8; ++w) t += sh[w][threadIdx.x];
    cov[(size_t)fk * 6 + threadIdx.x] = t * (1.f / (float)N_PIX);
  }
}

// ---------------------------------------------------------------------------
// Kernel 2: 2400 independent 3x3 symmetric eigendecompositions (closed form,
// f64 internally: covariances of N(0,1) data are near-isotropic).
// Outputs descending eigenvalues lam[3] and eigenvectors v0,v1,v2.
// ---------------------------------------------------------------------------
__device__ inline void eigvec3(double a, double b, double c, double d, double e,
                               double f, double lam, double v[3]) {
  double r0x = a - lam, r0y = b, r0z = c;
  double r1x = b, r1y = d - lam, r1z = e;
  double r2x = c, r2y = e, r2z = f - lam;
  double cx[3][3] = {
      {r0y * r1z - r0z * r1y, r0z * r1x - r0x * r1z, r0x * r1y - r0y * r1x},
      {r0y * r2z - r0z * r2y, r0z * r2x - r0x * r2z, r0x * r2y - r0y * r2x},
      {r1y * r2z - r1z * r2y, r1z * r2x - r1x * r2z, r1x * r2y - r1y * r2x}};
  int best = 0;
  double nb = cx[0][0] * cx[0][0] + cx[0][1] * cx[0][1] + cx[0][2] * cx[0][2];
  for (int i = 1; i < 3; ++i) {
    double n = cx[i][0] * cx[i][0] + cx[i][1] * cx[i][1] + cx[i][2] * cx[i][2];
    if (n > nb) { nb = n; best = i; }
  }
  if (nb < 1e-30) { v[0] = 1.0; v[1] = 0.0; v[2] = 0.0; return; }
  double inv = 1.0 / sqrt(nb);
  v[0] = cx[best][0] * inv; v[1] = cx[best][1] * inv; v[2] = cx[best][2] * inv;
}

__global__ void eig_kernel(const float* __restrict__ cov, float* __restrict__ lam,
                           float* __restrict__ ev0, float* __restrict__ ev1,
                           float* __restrict__ ev2) {
  int i = blockIdx.x * blockDim.x + threadIdx.x;
  if (i >= B_DIM * K_DIM) return;
  const float* m = cov + (size_t)i * 6;
  double a = m[0], b = m[1], c = m[2], d = m[3], e = m[4], f = m[5];
  double q = (a + d + f) / 3.0;
  double aa = a - q, dd = d - q, ff = f - q;
  double p2 = aa * aa + dd * dd + ff * ff + 2.0 * (b * b + c * c + e * e);
  double l0, l1, l2, u0[3], u1[3], u2[3];
  if (p2 < 1e-24) {
    l0 = l1 = l2 = q;
    u0[0] = 1; u0[1] = 0; u0[2] = 0;
    u1[0] = 0; u1[1] = 1; u1[2] = 0;
    u2[0] = 0; u2[1] = 0; u2[2] = 1;
  } else {
    double p = sqrt(p2 / 6.0), ip = 1.0 / p;
    double b11 = aa * ip, b12 = b * ip, b13 = c * ip;
    double b22 = dd * ip, b23 = e * ip, b33 = ff * ip;
    double detB = b11 * (b22 * b33 - b23 * b23) - b12 * (b12 * b33 - b23 * b13) +
                  b13 * (b12 * b23 - b22 * b13);
    double r = 0.5 * detB;
    r = fmin(1.0, fmax(-1.0, r));
    double phi = acos(r) / 3.0;
    l0 = q + 2.0 * p * cos(phi);                       // largest
    l2 = q + 2.0 * p * cos(phi + 2.0943951023931953);  // smallest
    l1 = 3.0 * q - l0 - l2;
    eigvec3(a, b, c, d, e, f, l0, u0);
    eigvec3(a, b, c, d, e, f, l2, u2);
    // middle vector: orthogonal complement (sign is irrelevant downstream)
    u1[0] = u2[1] * u0[2] - u2[2] * u0[1];
    u1[1] = u2[2] * u0[0] - u2[0] * u0[2];
    u1[2] = u2[0] * u0[1] - u2[1] * u0[0];
    double n = u1[0] * u1[0] + u1[1] * u1[1] + u1[2] * u1[2];
    double inv = (n > 1e-30) ? (1.0 / sqrt(n)) : 0.0;
    u1[0] *= inv; u1[1] *= inv; u1[2] *= inv;
  }
  lam[(size_t)i * 3 + 0] = (float)l0;
  lam[(size_t)i * 3 + 1] = (float)l1;
  lam[(size_t)i * 3 + 2] = (float)l2;
#pragma unroll
  for (int cidx = 0; cidx < 3; ++cidx) {
    ev0[(size_t)i * 3 + cidx] = (float)u0[cidx];
    ev1[(size_t)i * 3 + cidx] = (float)u1[cidx];
    ev2[(size_t)i * 3 + cidx] = (float)u2[cidx];
  }
}

// ---------------------------------------------------------------------------
// Kernel 3: banded GEMM via V_WMMA_F32_16X16X4_F32.
//   G_e[b,t,w] = sum_c ev0[b,t,c] * ev1[b,w,c]   (K=3 padded to 4)
//   dot_de[b,w,l] = G_e[b, w+l, w] for 0 <= l < 30
// One wave32 per (band-offset, w-tile, batch); A is shared by the two WMMAs.
// A 16x4 layout: lanes 0-15 hold {K0,K1}, lanes 16-31 hold {K2,K3}.
// D 16x16 layout: VGPR r -> row r + 8*(lane>=16), col = lane & 15.
// ---------------------------------------------------------------------------
__global__ void wmma_band_kernel(const float* __restrict__ ev0,
                                 const float* __restrict__ ev1,
                                 const float* __restrict__ ev2,
                                 float* __restrict__ dot_de,
                                 float* __restrict__ dot_df) {
  const int diff = blockIdx.x;          // 0..2  (t-tile minus w-tile)
  const int jw = blockIdx.y;            // 0..16 (w tile)
  const int b = blockIdx.z;             // 0..7
  const int wBase = jw * 16;
  const int tBase = wBase + diff * 16;  // <= 288+... max 288? (256+32=288) < 300
  const int lane = threadIdx.x;         // 0..31
  const int half = lane >> 4;           // 0: K0/K1 ; 1: K2/K3(=0)
  const int idx = lane & 15;

  // A: rows t = tBase+idx of ev0 (B is symmetric layout with w rows of ev1/ev2)
  v2f a = {0.f, 0.f};
  {
    int t = tBase + idx;
    if (t < K_DIM) {
      const float* p = ev0 + ((size_t)b * K_DIM + t) * 3;
      a.x = half ? p[2] : p[0];
      a.y = half ? 0.f : p[1];
    }
  }
  v2f be = {0.f, 0.f}, bf = {0.f, 0.f};
  {
    int w = wBase + idx;
    if (w < W_DIM) {
      const float* pe = ev1 + ((size_t)b * K_DIM + w) * 3;
      const float* pf = ev2 + ((size_t)b * K_DIM + w) * 3;
      be.x = half ? pe[2] : pe[0];
      be.y = half ? 0.f : pe[1];
      bf.x = half ? pf[2] : pf[0];
      bf.y = half ? 0.f : pf[1];
    }
  }

  v8f cz = {};
  // 8 args: (neg_a, A, neg_b, B, c_mod, C, reuse_a, reuse_b)
  v8f de = __builtin_amdgcn_wmma_f32_16x16x4_f32(false, a, false, be, (short)0,
                                                 cz, false, false);
  v8f df = __builtin_amdgcn_wmma_f32_16x16x4_f32(false, a, false, bf, (short)0,
                                                 cz, false, false);

  const int ww = wBase + idx;  // output column = lane & 15
#pragma unroll
  for (int r = 0; r < 8; ++r) {
    int tt = tBase + r + half * 8;
    int l = tt - ww;
    if (ww < W_DIM && l >= 0 && l < L_WIN) {
      size_t o = ((size_t)b * W_DIM + ww) * L_WIN + l;
      dot_de[o] = de[r];
      dot_df[o] = df[r];
    }
  }
}

// ---------------------------------------------------------------------------
// Kernel 4: per-(b,w) window statistics + scatter-add into P.
// One wave32 per (b,w); lanes 0..29 cover the l-window; shuffle reductions.
// ---------------------------------------------------------------------------
__global__ void post_kernel(const float* __restrict__ lam,
                            const float* __restrict__ ev1,
                            const float* __restrict__ ev2,
                            const float* __restrict__ dot_de,
                            const float* __restrict__ dot_df,
                            float* __restrict__ P) {
  const int bw = blockIdx.x;            // b*W_DIM + w
  const int b = bw / W_DIM, w = bw % W_DIM;
  const int l = threadIdx.x;            // 0..31, active < 30
  const bool act = (l < L_WIN);

  const size_t fw = (size_t)b * K_DIM + w;
  const float lam1 = lam[fw * 3 + 1];
  const float lam2 = lam[fw * 3 + 2];
  const float e0 = ev1[fw * 3 + 0], e1 = ev1[fw * 3 + 1];
  const float f0 = ev2[fw * 3 + 0], f1 = ev2[fw * 3 + 1];

  float s0 = 0.f, s1 = 0.f;
  const int t = w + l;
  if (act) {
    const float lam0t = lam[((size_t)b * K_DIM + t) * 3 + 0];
    const size_t o = ((size_t)b * W_DIM + w) * L_WIN + l;
    const float c1 = sqrtf(lam0t / lam1) * dot_de[o];
    const float c2 = sqrtf(lam0t / lam2) * dot_df[o];
    s0 = c1 * e0 + c2 * f0;
    s1 = c1 * e1 + c2 * f1;
  }
  const float n = (float)L_WIN;
  const float sum0 = wave_sum32(s0), sq0 = wave_sum32(s0 * s0);
  const float sum1 = wave_sum32(s1), sq1 = wave_sum32(s1 * s1);
  const float var0 = (sq0 - sum0 * sum0 / n) / (n - 1.f);
  const float var1 = (sq1 - sum1 * sum1 / n) / (n - 1.f);
  const float std0 = sqrtf(fmaxf(var0, 0.f));
  const float std1 = sqrtf(fmaxf(var1, 0.f));
  float p = act ? (s0 - (std0 / std1) * s1) : 0.f;
  const float pmean = wave_sum32(p) / n;
  if (act) {
    p -= pmean;
    atomicAdd(&P[(size_t)b * K_DIM + t], p);
  }
}

// ---------------------------------------------------------------------------
// Host launcher
// ---------------------------------------------------------------------------
extern "C" void kernel_launch(void* const* d_in, const int* in_sizes, int n_in,
                              void* d_out, int out_size, void* d_ws,
                              size_t ws_size, hipStream_t stream) {
  const float* images = (const float*)d_in[0];  // (8,3,300,64,64) f32
  float* P = (float*)d_out;                     // (8,300) f32

  // Workspace layout (floats):
  float* ws = (float*)d_ws;
  float* cov = ws;                      // 2400*6  = 14400
  float* lam = cov + 14400;             // 2400*3  =  7200
  float* ev0 = lam + 7200;              //            7200
  float* ev1 = ev0 + 7200;              //            7200
  float* ev2 = ev1 + 7200;              //            7200
  float* dot_de = ev2 + 7200;           // 8*270*30 = 64800
  float* dot_df = dot_de + 64800;       //            64800

  const int nP = B_DIM * K_DIM;  // 2400
  zero_kernel<<<(nP + 255) / 256, 256, 0, stream>>>(P, nP);

  cov_kernel<<<B_DIM * K_DIM, 256, 0, stream>>>(images, cov);

  eig_kernel<<<(nP + 127) / 128, 128, 0, stream>>>(cov, lam, ev0, ev1, ev2);

  wmma_band_kernel<<<dim3(3, 17, B_DIM), 32, 0, stream>>>(ev0, ev1, ev2, dot_de,
                                                          dot_df);

  post_kernel<<<B_DIM * W_DIM, 32, 0, stream>>>(lam, ev1, ev2, dot_de, dot_df,
                                                P);
}